// BiLSTMCRF_65841848648064
// MI455X (gfx1250) — compile-verified
//
#include <hip/hip_runtime.h>
#include <hip/hip_bf16.h>
#include <math.h>

#define TAGS 10
#define START_TAG 8
#define STOP_TAG 9
#define EDIM 100
#define HDIM 64          // per-direction hidden
#define GDIM 256         // 4*HDIM gates
#define BATCH 64
#define SEQ 512
#define NEGINF (-1000.0f)
#define LB (SEQ * BATCH) // 32768 (time-major rows: row = t*BATCH + b)

typedef float v2f __attribute__((ext_vector_type(2)));
typedef float v8f __attribute__((ext_vector_type(8)));

// V_WMMA_F32_16X16X4_F32: A 16x4 f32 (2 VGPR), B 4x16 f32 (2 VGPR), C/D 16x16 f32 (8 VGPR)
__device__ __forceinline__ v8f wmma4(v2f a, v2f b, v8f c) {
  return __builtin_amdgcn_wmma_f32_16x16x4_f32(
      /*neg_a=*/false, a, /*neg_b=*/false, b,
      /*c_mod=*/(short)0, c, /*reuse_a=*/false, /*reuse_b=*/false);
}

__device__ __forceinline__ float sigm(float x) { return 1.0f / (1.0f + __expf(-x)); }

// ---------------------------------------------------------------------------
// Kernel 1: embedding gather + input projection for BOTH directions.
// P[d][row=t*B+b][j] = sum_e emb[b][t][e] * Wih[d][j][e] + bih[d][j] + bhh[d][j]
// One 16-row M-tile per block; 8 waves x 2 N-tiles x 2 dirs.
// ---------------------------------------------------------------------------
#define ASTR 104  // LDS stride for 100-wide A tile (even -> float2 aligned)
__global__ void k_embed_ingemm(const int* __restrict__ sent,
                               const float* __restrict__ etab,
                               const float* __restrict__ Wih_f,
                               const float* __restrict__ bih_f,
                               const float* __restrict__ bhh_f,
                               const float* __restrict__ Wih_b,
                               const float* __restrict__ bih_b,
                               const float* __restrict__ bhh_b,
                               float* __restrict__ Pf,
                               float* __restrict__ Pb) {
  __shared__ float lA[16 * ASTR];
  const int m0 = blockIdx.x * 16;
  const int tid = threadIdx.x;

  // gather 16 embedding rows (row -> (t,b) -> word id -> table row)
  for (int idx = tid; idx < 16 * EDIM; idx += 256) {
    int r = idx / EDIM, e = idx - r * EDIM;
    int row = m0 + r;
    int t = row >> 6, b = row & 63;
    int w = sent[b * SEQ + t];
    lA[r * ASTR + e] = etab[(long)w * EDIM + e];
  }
  __syncthreads();

  const int wave = tid >> 5, lane = tid & 31;
  const int mrow = lane & 15;          // A row index / B column index
  const int koff = (lane >> 4) * 2;    // K sub-offset per ISA fragment layout

  #pragma unroll
  for (int sel = 0; sel < 4; ++sel) {
    const int d = sel >> 1;
    const int n0 = (wave * 2 + (sel & 1)) * 16;
    const float* Wih = d ? Wih_b : Wih_f;
    const float* bih = d ? bih_b : bih_f;
    const float* bhh = d ? bhh_b : bhh_f;
    float* P = d ? Pb : Pf;

    v8f acc = {};
    const float* wrow = Wih + (long)(n0 + mrow) * EDIM;
    #pragma unroll
    for (int k = 0; k < 25; ++k) {   // K = 100 = 25 * 4
      v2f a = *(const v2f*)&lA[mrow * ASTR + 4 * k + koff];
      v2f bq = *(const v2f*)&wrow[4 * k + koff];
      acc = wmma4(a, bq, acc);
    }
    const int col = n0 + mrow;
    const float bias = bih[col] + bhh[col];
    #pragma unroll
    for (int r = 0; r < 8; ++r) {
      int row = r + ((lane >> 4) << 3);
      P[(long)(m0 + row) * GDIM + col] = acc[r] + bias;
    }
  }
}

// ---------------------------------------------------------------------------
// Kernel 2: serial LSTM recurrence. 8 workgroups = 2 dirs x 4 batch-blocks(16).
// Per step: gates[16x256] = h[16x64] @ Whh^T (WMMA) + pregates; elementwise.
// Whh B-fragments are hoisted into registers (loop-invariant over 512 steps,
// 64 VGPRs/lane -- occupancy is irrelevant with only 8 resident WGs).
// Next step's pregate tile is prefetched to hide the L2 stream latency.
// ---------------------------------------------------------------------------
#define HSTR 66   // hbuf stride: breaks the 64-bank conflict on M-strided reads
#define GSTR 260  // gbuf stride
__global__ void k_lstm(const float* __restrict__ Pf,
                       const float* __restrict__ Pb,
                       const float* __restrict__ Whh_f,
                       const float* __restrict__ Whh_b,
                       const float* __restrict__ h0,
                       const float* __restrict__ c0,
                       float* __restrict__ hs) {
  __shared__ float hbuf[16 * HSTR];
  __shared__ float cbuf[16 * HDIM];
  __shared__ float gbuf[16 * GSTR];
  const int dir = blockIdx.x >> 2;
  const int bblk = blockIdx.x & 3;
  const int tid = threadIdx.x;
  const float* P = dir ? Pb : Pf;
  const float* Whh = dir ? Whh_b : Whh_f;

  for (int idx = tid; idx < 16 * HDIM; idx += 256) {
    int b = idx >> 6, k = idx & 63;
    hbuf[b * HSTR + k] = h0[(dir * BATCH + bblk * 16 + b) * HDIM + k];
    cbuf[idx]          = c0[(dir * BATCH + bblk * 16 + b) * HDIM + k];
  }
  __syncthreads();

  const int wave = tid >> 5, lane = tid & 31;
  const int mrow = lane & 15;
  const int koff = (lane >> 4) * 2;

  // Hoist Whh B-fragments for this wave's 2 N-tiles into registers.
  v2f bf[2][16];
  #pragma unroll
  for (int nt2 = 0; nt2 < 2; ++nt2) {
    const int n0 = (wave * 2 + nt2) * 16;
    const float* wrow = Whh + (long)(n0 + mrow) * HDIM;
    #pragma unroll
    for (int k = 0; k < 16; ++k)
      bf[nt2][k] = *(const v2f*)&wrow[4 * k + koff];
  }

  for (int step = 0; step < SEQ; ++step) {
    const int t = dir ? (SEQ - 1 - step) : step;
    const float* prow = P + ((long)t * BATCH + bblk * 16) * GDIM;

    // Prefetch next step's 16x256 pregate tile (256 threads x 64B lines).
    if (step + 1 < SEQ) {
      const int tn = dir ? (SEQ - 2 - step) : (step + 1);
      const float* pnext = P + ((long)tn * BATCH + bblk * 16) * GDIM;
      __builtin_prefetch(pnext + tid * 16, 0, 0);
    }

    #pragma unroll
    for (int nt2 = 0; nt2 < 2; ++nt2) {
      const int n0 = (wave * 2 + nt2) * 16;
      v8f acc = {};
      #pragma unroll
      for (int k = 0; k < 16; ++k) {  // K = 64 = 16 * 4
        v2f a = *(const v2f*)&hbuf[mrow * HSTR + 4 * k + koff];
        acc = wmma4(a, bf[nt2][k], acc);
      }
      const int col = n0 + mrow;
      #pragma unroll
      for (int r = 0; r < 8; ++r) {
        int row = r + ((lane >> 4) << 3);
        gbuf[row * GSTR + col] = acc[r] + prow[(long)row * GDIM + col];
      }
    }
    __syncthreads();

    for (int idx = tid; idx < 16 * HDIM; idx += 256) {
      int b = idx >> 6, k = idx & 63;
      float gi = gbuf[b * GSTR + k];
      float gf = gbuf[b * GSTR + 64 + k];
      float gg = gbuf[b * GSTR + 128 + k];
      float go = gbuf[b * GSTR + 192 + k];
      float cc = sigm(gf) * cbuf[idx] + sigm(gi) * tanhf(gg);
      float hh = sigm(go) * tanhf(cc);
      cbuf[idx] = cc;
      hbuf[b * HSTR + k] = hh;
      hs[((long)t * BATCH + bblk * 16 + b) * (2 * HDIM) + dir * HDIM + k] = hh;
    }
    __syncthreads();
  }
}

// ---------------------------------------------------------------------------
// Kernel 3: output projection. logits[b][t][j<16] = hs[row] @ Wout^T + bout.
// N padded 10->16 (pad columns multiply by zero). One M-tile per wave.
// ---------------------------------------------------------------------------
__global__ void k_proj(const float* __restrict__ hs,
                       const float* __restrict__ Wout,
                       const float* __restrict__ bout,
                       float* __restrict__ logits) {
  const int tid = threadIdx.x;
  const int wave = tid >> 5, lane = tid & 31;
  const int m0 = (blockIdx.x * 8 + wave) * 16;
  const int mrow = lane & 15;
  const int koff = (lane >> 4) * 2;
  const bool valid = mrow < TAGS;

  const float* arow = hs + (long)(m0 + mrow) * (2 * HDIM);
  const float* wrow = Wout + (long)mrow * (2 * HDIM);
  v8f acc = {};
  #pragma unroll
  for (int k = 0; k < 32; ++k) {  // K = 128 = 32 * 4
    v2f a = *(const v2f*)&arow[4 * k + koff];
    v2f bq;
    if (valid) { bq = *(const v2f*)&wrow[4 * k + koff]; }
    else       { bq.x = 0.0f; bq.y = 0.0f; }
    acc = wmma4(a, bq, acc);   // full EXEC here (divergence reconverged)
  }
  const float bias = valid ? bout[mrow] : 0.0f;
  #pragma unroll
  for (int r = 0; r < 8; ++r) {
    int row = m0 + r + ((lane >> 4) << 3);  // row = t*BATCH + b
    int t = row >> 6, b = row & 63;
    logits[((long)b * SEQ + t) * 16 + mrow] = acc[r] + bias;
  }
}

// ---------------------------------------------------------------------------
// Kernel 4: CRF forward scan + backtrack. One wave32 per batch sequence;
// tag j lives in lane j (T=10), alpha broadcast via wave shuffles.
// ---------------------------------------------------------------------------
__global__ void k_viterbi(const float* __restrict__ logits,
                          const float* __restrict__ trans,
                          int* __restrict__ bptrs,
                          float* __restrict__ out) {
  const int b = blockIdx.x;
  const int j = threadIdx.x;     // 32 lanes; lanes >= TAGS idle-carry NEGINF

  float tr[TAGS];
  #pragma unroll
  for (int i = 0; i < TAGS; ++i)
    tr[i] = (j < TAGS) ? trans[i * TAGS + j] : NEGINF;

  float alpha = (j == START_TAG) ? 0.0f : NEGINF;

  for (int t = 0; t < SEQ; ++t) {
    float frame = (j < TAGS) ? logits[((long)b * SEQ + t) * 16 + j] : NEGINF;
    float m = -3.4e38f; int arg = 0;
    float s[TAGS];
    #pragma unroll
    for (int i = 0; i < TAGS; ++i) {
      float ai = __shfl(alpha, i, 32);
      float v = ai + tr[i];
      s[i] = v;
      if (v > m) { m = v; arg = i; }
    }
    float sum = 0.0f;
    #pragma unroll
    for (int i = 0; i < TAGS; ++i) sum += __expf(s[i] - m);
    if (j < TAGS) bptrs[((long)b * SEQ + t) * TAGS + j] = arg;
    float na = frame + m + __logf(sum);
    alpha = (j < TAGS) ? na : NEGINF;
  }

  float fin = (j < TAGS) ? (alpha + trans[j * TAGS + STOP_TAG]) : NEGINF;
  float m = -3.4e38f; int best = 0;
  float vals[TAGS];
  #pragma unroll
  for (int i = 0; i < TAGS; ++i) {
    float v = __shfl(fin, i, 32);
    vals[i] = v;
    if (v > m) { m = v; best = i; }
  }
  float sum = 0.0f;
  #pragma unroll
  for (int i = 0; i < TAGS; ++i) sum += __expf(vals[i] - m);
  float score = m + __logf(sum);

  if (j == 0) {
    out[b] = score;
    float* path = out + BATCH + (long)b * SEQ;
    int cur = best;
    path[SEQ - 1] = (float)cur;
    for (int t = SEQ - 1; t >= 1; --t) {
      cur = bptrs[((long)b * SEQ + t) * TAGS + cur];
      path[t - 1] = (float)cur;
    }
  }
}

// ---------------------------------------------------------------------------
extern "C" void kernel_launch(void* const* d_in, const int* in_sizes, int n_in,
                              void* d_out, int out_size, void* d_ws, size_t ws_size,
                              hipStream_t stream) {
  const int*   sent  = (const int*)  d_in[0];
  const float* etab  = (const float*)d_in[1];
  const float* Wih_f = (const float*)d_in[2];
  const float* Whh_f = (const float*)d_in[3];
  const float* bih_f = (const float*)d_in[4];
  const float* bhh_f = (const float*)d_in[5];
  const float* Wih_b = (const float*)d_in[6];
  const float* Whh_b = (const float*)d_in[7];
  const float* bih_b = (const float*)d_in[8];
  const float* bhh_b = (const float*)d_in[9];
  const float* Wout  = (const float*)d_in[10];
  const float* bout  = (const float*)d_in[11];
  const float* trans = (const float*)d_in[12];
  const float* h0    = (const float*)d_in[13];
  const float* c0    = (const float*)d_in[14];

  float* Pf     = (float*)d_ws;                       // [LB, 256]
  float* Pb     = Pf + (size_t)LB * GDIM;             // [LB, 256]
  float* hs     = Pb + (size_t)LB * GDIM;             // [LB, 128]
  float* logits = hs + (size_t)LB * (2 * HDIM);       // [B, L, 16]
  int*   bptrs  = (int*)(logits + (size_t)LB * 16);   // [B, L, 10]
  float* out    = (float*)d_out;                      // 64 scores + 64*512 path

  k_embed_ingemm<<<LB / 16, 256, 0, stream>>>(sent, etab,
                                              Wih_f, bih_f, bhh_f,
                                              Wih_b, bih_b, bhh_b, Pf, Pb);
  k_lstm<<<8, 256, 0, stream>>>(Pf, Pb, Whh_f, Whh_b, h0, c0, hs);
  k_proj<<<LB / 16 / 8, 256, 0, stream>>>(hs, Wout, bout, logits);
  k_viterbi<<<BATCH, 32, 0, stream>>>(logits, trans, bptrs, out);
}